// MatrixExponential_48103633715412
// MI455X (gfx1250) — compile-verified
//
#include <hip/hip_runtime.h>
#include <math.h>

// CDNA5 / gfx1250: wave32, V_WMMA_F32_16X16X4_F32 (A:16x4 f32 = v2f per lane,
// B:4x16 f32 = v2f per lane, C/D:16x16 f32 = v8f per lane).
typedef float v2f __attribute__((ext_vector_type(2)));
typedef float v8f __attribute__((ext_vector_type(8)));

#define LDM 65           // padded row stride for 64x64 LDS matrices (bank-friendly)
#define NWAVES 8
#define BLOCK 256

// Pade-7 coefficients (scipy/jax expm, order 7)
#define B0 17297280.0f
#define B1 8648640.0f
#define B2 1995840.0f
#define B3 277200.0f
#define B4 25200.0f
#define B5 1512.0f
#define B6 56.0f
#define B7 1.0f
#define THETA7 3.925724783138660f

__device__ __forceinline__ v8f wmma_f32(v2f a, v2f b, v8f c) {
  return __builtin_amdgcn_wmma_f32_16x16x4_f32(
      /*neg_a=*/false, a, /*neg_b=*/false, b,
      /*c_mod=*/(short)0, c, /*reuse_a=*/false, /*reuse_b=*/false);
}

// Compute the wave's two 16x16 C tiles (tile row tr, tile cols tc0, tc0+1)
// of C = A*B where A,B are 64x64 (stride LDM) in LDS. Result in fragments.
__device__ __forceinline__ void wave_mm_frag(const float* A, const float* B,
                                             v8f acc[2], int tr, int tc0,
                                             int half, int m) {
  v8f z = {};
  acc[0] = z;
  acc[1] = z;
  const int arow = (tr * 16 + m) * LDM;
  const int bc0 = tc0 * 16 + m;
  const int bc1 = bc0 + 16;
#pragma unroll
  for (int kb = 0; kb < 16; ++kb) {
    int k = kb * 4 + 2 * half;  // this lane's two K values: k, k+1
    v2f a, b0, b1;
    a.x = A[arow + k];
    a.y = A[arow + k + 1];
    b0.x = B[k * LDM + bc0];
    b0.y = B[(k + 1) * LDM + bc0];
    b1.x = B[k * LDM + bc1];
    b1.y = B[(k + 1) * LDM + bc1];
    acc[0] = wmma_f32(a, b0, acc[0]);
    acc[1] = wmma_f32(a, b1, acc[1]);
  }
}

__device__ __forceinline__ void wave_store_frag(float* D, const v8f acc[2],
                                                int tr, int tc0, int half,
                                                int m) {
#pragma unroll
  for (int t = 0; t < 2; ++t) {
    int col = (tc0 + t) * 16 + m;
#pragma unroll
    for (int r = 0; r < 8; ++r) {
      int row = tr * 16 + 8 * half + r;
      D[row * LDM + col] = acc[t][r];
    }
  }
}

__device__ __forceinline__ void wave_mm_store(float* D, const float* A,
                                              const float* B, int tr, int tc0,
                                              int half, int m) {
  v8f acc[2];
  wave_mm_frag(A, B, acc, tr, tc0, half, m);
  wave_store_frag(D, acc, tr, tc0, half, m);
}

__global__ __launch_bounds__(BLOCK, 1) void MatrixExponential_48103633715412_kernel(
    const float* __restrict__ X, float* __restrict__ Out, int nbatch) {
  __shared__ float sA[64 * LDM];  // scaled A, later squaring ping buffer
  __shared__ float sB[64 * LDM];  // A2, later P -> solution X
  __shared__ float sC[64 * LDM];  // A4, later W, later Q
  __shared__ float sRed[64];
  __shared__ float sScale, sPivVal;
  __shared__ int sSteps, sPiv;

  const int tid = threadIdx.x;
  const int lane = tid & 31;
  const int wave = tid >> 5;
  const int half = lane >> 4;   // 0: lanes 0-15, 1: lanes 16-31
  const int m = lane & 15;      // row (A) / col (B,C) index within tile
  const int tr = wave >> 1;     // wave's tile row   (0..3)
  const int tc0 = (wave & 1) * 2;  // wave's first tile col (0 or 2)

  const int b = blockIdx.x;
  if (b >= nbatch) return;
  const float* gin = X + (size_t)b * 4096;
  float* gout = Out + (size_t)b * 4096;

  // ---- load A into LDS (coalesced) ----
#pragma unroll
  for (int j = 0; j < 16; ++j) {
    int idx = j * BLOCK + tid;
    sA[(idx >> 6) * LDM + (idx & 63)] = gin[idx];
  }
  __syncthreads();

  // ---- 1-norm (max column abs-sum), pick scaling s ----
  if (tid < 64) {
    float ssum = 0.0f;
    for (int r = 0; r < 64; ++r) ssum += fabsf(sA[r * LDM + tid]);
    sRed[tid] = ssum;
  }
  __syncthreads();
  if (tid == 0) {
    float mx = 0.0f;
    for (int i = 0; i < 64; ++i) mx = fmaxf(mx, sRed[i]);
    int s = 0;
    if (mx > THETA7 && mx < 3.0e37f) {
      s = (int)ceilf(log2f(mx / THETA7));
      if (s < 0) s = 0;
      if (s > 24) s = 24;
    }
    sSteps = s;
    sScale = exp2f(-(float)s);
  }
  __syncthreads();
  const int nsq = sSteps;
  const float scale = sScale;
#pragma unroll
  for (int j = 0; j < 16; ++j) {
    int idx = j * BLOCK + tid;
    sA[(idx >> 6) * LDM + (idx & 63)] *= scale;
  }
  __syncthreads();

  // ---- A2 = A*A -> sB ----
  wave_mm_store(sB, sA, sA, tr, tc0, half, m);
  __syncthreads();
  // ---- A4 = A2*A2 -> sC ----
  wave_mm_store(sC, sB, sB, tr, tc0, half, m);
  __syncthreads();

  // ---- A6 = A2*A4 (fragments only) ----
  v8f a6[2];
  wave_mm_frag(sB, sC, a6, tr, tc0, half, m);

  // ---- V = b6*A6 + b4*A4 + b2*A2 + b0*I ; W = b7*A6 + b5*A4 + b3*A2 + b1*I ----
  v8f vf[2], wf[2];
#pragma unroll
  for (int t = 0; t < 2; ++t) {
    int col = (tc0 + t) * 16 + m;
#pragma unroll
    for (int r = 0; r < 8; ++r) {
      int row = tr * 16 + 8 * half + r;
      float a2v = sB[row * LDM + col];
      float a4v = sC[row * LDM + col];
      float dg = (row == col) ? 1.0f : 0.0f;
      vf[t][r] = B6 * a6[t][r] + B4 * a4v + B2 * a2v + B0 * dg;
      wf[t][r] = B7 * a6[t][r] + B5 * a4v + B3 * a2v + B1 * dg;
    }
  }
  __syncthreads();            // all reads of sC (A4) done
  wave_store_frag(sC, wf, tr, tc0, half, m);  // sC <- W
  __syncthreads();

  // ---- U = A*W (fragments), then P = V+U -> sB, Q = V-U -> sC ----
  v8f uf[2];
  wave_mm_frag(sA, sC, uf, tr, tc0, half, m);
  v8f pf[2];
  pf[0] = vf[0] + uf[0];
  pf[1] = vf[1] + uf[1];
  wave_store_frag(sB, pf, tr, tc0, half, m);  // sB <- P (nobody reads sB now)
  __syncthreads();            // all mm_frag reads of sC done
  v8f qf[2];
  qf[0] = vf[0] - uf[0];
  qf[1] = vf[1] - uf[1];
  wave_store_frag(sC, qf, tr, tc0, half, m);  // sC <- Q
  __syncthreads();

  // ---- Gauss-Jordan with partial pivoting: solve Q X = P, X -> sB ----
  for (int k = 0; k < 64; ++k) {
    if (tid < 64) sRed[tid] = (tid >= k) ? fabsf(sC[tid * LDM + k]) : -1.0f;
    __syncthreads();
    if (tid == 0) {
      int p = k;
      float bv = sRed[k];
      for (int i = k + 1; i < 64; ++i)
        if (sRed[i] > bv) { bv = sRed[i]; p = i; }
      sPiv = p;
    }
    __syncthreads();
    int p = sPiv;
    if (p != k) {
      if (tid < 64) {
        float t1 = sC[k * LDM + tid];
        sC[k * LDM + tid] = sC[p * LDM + tid];
        sC[p * LDM + tid] = t1;
      } else if (tid < 128) {
        int c = tid - 64;
        float t1 = sB[k * LDM + c];
        sB[k * LDM + c] = sB[p * LDM + c];
        sB[p * LDM + c] = t1;
      }
    }
    __syncthreads();
    if (tid == 0) sPivVal = 1.0f / sC[k * LDM + k];
    __syncthreads();
    float inv = sPivVal;
    if (tid < 64) sC[k * LDM + tid] *= inv;
    else if (tid < 128) sB[k * LDM + (tid - 64)] *= inv;
    __syncthreads();
    {
      int row = tid >> 2;       // 4 threads per row, all in the same wave
      int g = tid & 3;          // each covers 32 of the 128 augmented columns
      if (row != k) {
        float f = sC[row * LDM + k];
#pragma unroll
        for (int cc = 0; cc < 32; ++cc) {
          int c = g * 32 + cc;
          if (c < 64)
            sC[row * LDM + c] -= f * sC[k * LDM + c];
          else {
            int c2 = c - 64;
            sB[row * LDM + c2] -= f * sB[k * LDM + c2];
          }
        }
      }
    }
    __syncthreads();
  }
  // solution X now in sB

  // ---- squaring phase: X <- X^2, nsq times (ping-pong sB <-> sA) ----
  float* cur = sB;
  float* nxt = sA;
  for (int i = 0; i < nsq; ++i) {
    wave_mm_store(nxt, cur, cur, tr, tc0, half, m);
    __syncthreads();
    float* t = cur;
    cur = nxt;
    nxt = t;
  }

  // ---- write result ----
#pragma unroll
  for (int j = 0; j < 16; ++j) {
    int idx = j * BLOCK + tid;
    gout[idx] = cur[(idx >> 6) * LDM + (idx & 63)];
  }
}

extern "C" void kernel_launch(void* const* d_in, const int* in_sizes, int n_in,
                              void* d_out, int out_size, void* d_ws, size_t ws_size,
                              hipStream_t stream) {
  const float* X = (const float*)d_in[0];
  float* Out = (float*)d_out;
  int nbatch = in_sizes[0] / 4096;  // 8192 batches of 64x64
  MatrixExponential_48103633715412_kernel<<<nbatch, BLOCK, 0, stream>>>(X, Out, nbatch);
}